// RestGCN_1597727834504
// MI455X (gfx1250) — compile-verified
//
#include <hip/hip_runtime.h>

#define N_NODES   10000
#define N_EDGES   320000
#define HIDDEN    256
#define N_CLASSES 10
#define N_GRAPHS  64

typedef __attribute__((ext_vector_type(2))) float v2f;
typedef __attribute__((ext_vector_type(8))) float v8f;

// ---------------------------------------------------------------- init
__global__ void init_kernel(float* deg, int* cursor, float* pooled, float* counts) {
    int i = blockIdx.x * blockDim.x + threadIdx.x;
    if (i < N_NODES) { deg[i] = 1.0f; cursor[i] = 0; }
    if (i < N_GRAPHS * HIDDEN) pooled[i] = 0.0f;
    if (i < N_GRAPHS) counts[i] = 0.0f;
}

// ------------------------------------------------- degree (in-deg + self)
__global__ void deg_kernel(const int* __restrict__ dst, float* deg) {
    int e = blockIdx.x * blockDim.x + threadIdx.x;
    if (e < N_EDGES) atomicAdd(&deg[dst[e]], 1.0f);
}

__global__ void dis_kernel(const float* __restrict__ deg, float* dis) {
    int n = blockIdx.x * blockDim.x + threadIdx.x;
    if (n < N_NODES) dis[n] = rsqrtf(deg[n]);
}

// ------------------------------------------- exclusive scan of in-degrees
__global__ void scan_kernel(const float* __restrict__ deg, int* rowoff) {
    __shared__ int lds[1024];
    constexpr int CH = (N_NODES + 1023) / 1024; // 10
    int t = threadIdx.x;
    int base = t * CH;
    int local[CH];
    int sum = 0;
    for (int i = 0; i < CH; ++i) {
        int idx = base + i;
        int v = 0;
        if (idx < N_NODES) v = (int)(deg[idx] + 0.5f) - 1; // in-degree
        local[i] = v;
        sum += v;
    }
    lds[t] = sum;
    __syncthreads();
    for (int off = 1; off < 1024; off <<= 1) {
        int v = (t >= off) ? lds[t - off] : 0;
        __syncthreads();
        lds[t] += v;
        __syncthreads();
    }
    int run = (t == 0) ? 0 : lds[t - 1];
    for (int i = 0; i < CH; ++i) {
        int idx = base + i;
        if (idx < N_NODES) rowoff[idx] = run;
        run += local[i];
    }
    if (t == 1023) rowoff[N_NODES] = lds[1023];
}

// --------------------------------------------------------------- CSR fill
__global__ void csr_fill_kernel(const int* __restrict__ src, const int* __restrict__ dst,
                                const int* __restrict__ rowoff, int* cursor,
                                const float* __restrict__ dis,
                                int* csr_src, float* csr_w) {
    int e = blockIdx.x * blockDim.x + threadIdx.x;
    if (e >= N_EDGES) return;
    int s = src[e], d = dst[e];
    int idx = rowoff[d] + atomicAdd(&cursor[d], 1);
    csr_src[idx] = s;
    csr_w[idx] = dis[s] * dis[d];
}

// ----------------------------------------------- WMMA GEMM: C = (A[+A2])@W
// One wave computes a 32x64 output strip using V_WMMA_F32_16X16X4_F32:
// two 16-row blocks (m0, m1) x four 16-col blocks, 8 accumulators.
template <bool HAS_A2>
__global__ void gemm_wmma_kernel(const float* __restrict__ A, const float* __restrict__ A2,
                                 const float* __restrict__ W, float* __restrict__ C) {
    constexpr int NT    = HIDDEN / 64;            // 4 n-strips of 64
    constexpr int MT    = (N_NODES + 31) / 32;    // 313 m-strips of 32
    constexpr int TILES = MT * NT;                // 1252 waves
    int wid  = (blockIdx.x * blockDim.x + threadIdx.x) >> 5;
    int lane = threadIdx.x & 31;
    if (wid >= TILES) return;

    int m0 = (wid / NT) * 32;
    int n0 = (wid % NT) * 64;
    bool two = (m0 + 16) < N_NODES;        // wave-uniform tail predicate
    int m1 = two ? (m0 + 16) : m0;         // clamp: tail wave computes (discarded) dup
    int l15   = lane & 15;
    int khalf = (lane >> 4) * 2;           // 0 or 2

    const float* arow0  = A + (size_t)(m0 + l15) * HIDDEN;
    const float* arow1  = A + (size_t)(m1 + l15) * HIDDEN;
    const float* arow0r = HAS_A2 ? (A2 + (size_t)(m0 + l15) * HIDDEN) : nullptr;
    const float* arow1r = HAS_A2 ? (A2 + (size_t)(m1 + l15) * HIDDEN) : nullptr;

    v8f acc[8];
#pragma unroll
    for (int j = 0; j < 8; ++j) acc[j] = (v8f){0.f,0.f,0.f,0.f,0.f,0.f,0.f,0.f};

    for (int kb = 0; kb < HIDDEN / 4; ++kb) {
        int k = kb * 4 + khalf;
        v2f a0 = *(const v2f*)(arow0 + k);
        v2f a1 = *(const v2f*)(arow1 + k);
        if (HAS_A2) {
            a0 += *(const v2f*)(arow0r + k);
            a1 += *(const v2f*)(arow1r + k);
        }
        const float* w0 = W + (size_t)k * HIDDEN + n0 + l15;
#pragma unroll
        for (int j = 0; j < 4; ++j) {
            v2f b;
            b.x = w0[j * 16];            // row k
            b.y = w0[HIDDEN + j * 16];   // row k+1
            acc[j] = __builtin_amdgcn_wmma_f32_16x16x4_f32(
                false, a0, false, b, (short)0, acc[j], false, false);
            acc[4 + j] = __builtin_amdgcn_wmma_f32_16x16x4_f32(
                false, a1, false, b, (short)0, acc[4 + j], false, false);
        }
    }

    int rhalf  = 8 * (lane >> 4);
    int nb     = n0 + l15;
#pragma unroll
    for (int j = 0; j < 4; ++j)
#pragma unroll
        for (int r = 0; r < 8; ++r)
            C[(size_t)(m0 + rhalf + r) * HIDDEN + nb + j * 16] = acc[j][r];
    if (two) {
#pragma unroll
        for (int j = 0; j < 4; ++j)
#pragma unroll
            for (int r = 0; r < 8; ++r)
                C[(size_t)(m1 + rhalf + r) * HIDDEN + nb + j * 16] = acc[4 + j][r];
    }
}

// -------------------------------- CSR aggregation + self-loop + bias (+relu)
// One wave per node; lane owns 8 contiguous channels (two float4s).
// Edge loop unrolled x2 to keep two 1KB row-gathers in flight.
__global__ void aggregate_kernel(const float* __restrict__ hW,
                                 const int* __restrict__ rowoff,
                                 const int* __restrict__ csr_src,
                                 const float* __restrict__ csr_w,
                                 const float* __restrict__ deg,
                                 const float* __restrict__ bias,
                                 float* __restrict__ out, int relu) {
    int wid  = (blockIdx.x * blockDim.x + threadIdx.x) >> 5;
    int lane = threadIdx.x & 31;
    if (wid >= N_NODES) return;
    int c0 = lane * 8;

    float acc[8] = {0.f,0.f,0.f,0.f,0.f,0.f,0.f,0.f};
    int beg = rowoff[wid], end = rowoff[wid + 1];
    int e = beg;
    for (; e + 1 < end; e += 2) {
        int   s0 = csr_src[e],     s1 = csr_src[e + 1];
        float w0 = csr_w[e],       w1 = csr_w[e + 1];
        const float4* r0 = (const float4*)(hW + (size_t)s0 * HIDDEN + c0);
        const float4* r1 = (const float4*)(hW + (size_t)s1 * HIDDEN + c0);
        float4 p0 = r0[0], q0 = r0[1];
        float4 p1 = r1[0], q1 = r1[1];
        acc[0] += w0 * p0.x; acc[1] += w0 * p0.y; acc[2] += w0 * p0.z; acc[3] += w0 * p0.w;
        acc[4] += w0 * q0.x; acc[5] += w0 * q0.y; acc[6] += w0 * q0.z; acc[7] += w0 * q0.w;
        acc[0] += w1 * p1.x; acc[1] += w1 * p1.y; acc[2] += w1 * p1.z; acc[3] += w1 * p1.w;
        acc[4] += w1 * q1.x; acc[5] += w1 * q1.y; acc[6] += w1 * q1.z; acc[7] += w1 * q1.w;
    }
    if (e < end) {
        int   s = csr_src[e];
        float w = csr_w[e];
        const float4* r = (const float4*)(hW + (size_t)s * HIDDEN + c0);
        float4 p = r[0], q = r[1];
        acc[0] += w * p.x; acc[1] += w * p.y; acc[2] += w * p.z; acc[3] += w * p.w;
        acc[4] += w * q.x; acc[5] += w * q.y; acc[6] += w * q.z; acc[7] += w * q.w;
    }
    float self = 1.0f / deg[wid];   // dis^2
    const float4* r = (const float4*)(hW + (size_t)wid * HIDDEN + c0);
    float4 p = r[0], q = r[1];
    float v[8];
    v[0] = acc[0] + self * p.x + bias[c0 + 0];
    v[1] = acc[1] + self * p.y + bias[c0 + 1];
    v[2] = acc[2] + self * p.z + bias[c0 + 2];
    v[3] = acc[3] + self * p.w + bias[c0 + 3];
    v[4] = acc[4] + self * q.x + bias[c0 + 4];
    v[5] = acc[5] + self * q.y + bias[c0 + 5];
    v[6] = acc[6] + self * q.z + bias[c0 + 6];
    v[7] = acc[7] + self * q.w + bias[c0 + 7];
    if (relu)
#pragma unroll
        for (int i = 0; i < 8; ++i) v[i] = fmaxf(v[i], 0.f);
    float4* o = (float4*)(out + (size_t)wid * HIDDEN + c0);
    o[0] = make_float4(v[0], v[1], v[2], v[3]);
    o[1] = make_float4(v[4], v[5], v[6], v[7]);
}

// ---------------------------------------------------------------- pooling
__global__ void counts_kernel(const int* __restrict__ batch, float* counts) {
    int n = blockIdx.x * blockDim.x + threadIdx.x;
    if (n < N_NODES) atomicAdd(&counts[batch[n]], 1.0f);
}

__global__ void pool_kernel(const float* __restrict__ x6, const int* __restrict__ batch,
                            float* pooled) {
    int wid  = (blockIdx.x * blockDim.x + threadIdx.x) >> 5;
    int lane = threadIdx.x & 31;
    if (wid >= N_NODES) return;
    int g = batch[wid];
    const float* row = x6 + (size_t)wid * HIDDEN;
    int c0 = lane * 8;
#pragma unroll
    for (int i = 0; i < 8; ++i) atomicAdd(&pooled[g * HIDDEN + c0 + i], row[c0 + i]);
}

// ------------------------------------------------------------ final linear
__global__ void final_kernel(const float* __restrict__ pooled, const float* __restrict__ counts,
                             const float* __restrict__ linW, const float* __restrict__ linb,
                             float* __restrict__ out) {
    int i = blockIdx.x * blockDim.x + threadIdx.x;
    if (i >= N_GRAPHS * N_CLASSES) return;
    int g = i / N_CLASSES, c = i % N_CLASSES;
    float inv = 1.0f / fmaxf(counts[g], 1.0f);
    float s = linb[c];
    for (int k = 0; k < HIDDEN; ++k)
        s += pooled[g * HIDDEN + k] * inv * linW[k * N_CLASSES + c];
    out[i] = s;
}

// ================================================================ launcher
extern "C" void kernel_launch(void* const* d_in, const int* in_sizes, int n_in,
                              void* d_out, int out_size, void* d_ws, size_t ws_size,
                              hipStream_t stream) {
    const float* x     = (const float*)d_in[0];
    const float* convW = (const float*)d_in[1];
    const float* convB = (const float*)d_in[2];
    const float* linW  = (const float*)d_in[3];
    const float* linb  = (const float*)d_in[4];
    const int*   eidx  = (const int*)d_in[5];
    const int*   batch = (const int*)d_in[6];
    const int* src = eidx;
    const int* dst = eidx + N_EDGES;
    float* out = (float*)d_out;

    // carve workspace
    char* base = (char*)d_ws;
    size_t off = 0;
    auto alloc = [&](size_t bytes) -> void* {
        void* p = base + off;
        off = (off + bytes + 255) & ~(size_t)255;
        return p;
    };
    float* deg     = (float*)alloc(N_NODES * sizeof(float));
    float* dis     = (float*)alloc(N_NODES * sizeof(float));
    int*   rowoff  = (int*)  alloc((N_NODES + 1) * sizeof(int));
    int*   cursor  = (int*)  alloc(N_NODES * sizeof(int));
    int*   csr_src = (int*)  alloc(N_EDGES * sizeof(int));
    float* csr_w   = (float*)alloc(N_EDGES * sizeof(float));
    float* pooled  = (float*)alloc(N_GRAPHS * HIDDEN * sizeof(float));
    float* counts  = (float*)alloc(N_GRAPHS * sizeof(float));
    float* h[7];
    for (int i = 0; i < 7; ++i) h[i] = (float*)alloc((size_t)N_NODES * HIDDEN * sizeof(float));
    float* hW = h[0];
    float* x1 = h[1]; float* x2 = h[2]; float* x3 = h[3];
    float* x4 = h[4]; float* x5 = h[5]; float* x6 = h[6];

    const int B = 256;
    // graph preprocessing
    init_kernel<<<64, B, 0, stream>>>(deg, cursor, pooled, counts);
    deg_kernel<<<(N_EDGES + B - 1) / B, B, 0, stream>>>(dst, deg);
    dis_kernel<<<(N_NODES + B - 1) / B, B, 0, stream>>>(deg, dis);
    scan_kernel<<<1, 1024, 0, stream>>>(deg, rowoff);
    csr_fill_kernel<<<(N_EDGES + B - 1) / B, B, 0, stream>>>(src, dst, rowoff, cursor,
                                                             dis, csr_src, csr_w);

    constexpr int TILES = ((N_NODES + 31) / 32) * (HIDDEN / 64);   // 1252 waves
    const int gemm_blocks = (TILES * 32 + B - 1) / B;
    const int agg_blocks  = (N_NODES * 32 + B - 1) / B;

    // L0: x -> x1          (no residual)
    gemm_wmma_kernel<false><<<gemm_blocks, B, 0, stream>>>(x, nullptr, convW + 0 * HIDDEN * HIDDEN, hW);
    aggregate_kernel<<<agg_blocks, B, 0, stream>>>(hW, rowoff, csr_src, csr_w, deg,
                                                   convB + 0 * HIDDEN, x1, 1);
    // L1: x1 -> x2         (no residual)
    gemm_wmma_kernel<false><<<gemm_blocks, B, 0, stream>>>(x1, nullptr, convW + 1 * HIDDEN * HIDDEN, hW);
    aggregate_kernel<<<agg_blocks, B, 0, stream>>>(hW, rowoff, csr_src, csr_w, deg,
                                                   convB + 1 * HIDDEN, x2, 1);
    // L2: x2+x1 -> x3
    gemm_wmma_kernel<true><<<gemm_blocks, B, 0, stream>>>(x2, x1, convW + 2 * HIDDEN * HIDDEN, hW);
    aggregate_kernel<<<agg_blocks, B, 0, stream>>>(hW, rowoff, csr_src, csr_w, deg,
                                                   convB + 2 * HIDDEN, x3, 1);
    // L3: x3+x2 -> x4
    gemm_wmma_kernel<true><<<gemm_blocks, B, 0, stream>>>(x3, x2, convW + 3 * HIDDEN * HIDDEN, hW);
    aggregate_kernel<<<agg_blocks, B, 0, stream>>>(hW, rowoff, csr_src, csr_w, deg,
                                                   convB + 3 * HIDDEN, x4, 1);
    // L4: x3+x4 -> x5
    gemm_wmma_kernel<true><<<gemm_blocks, B, 0, stream>>>(x3, x4, convW + 4 * HIDDEN * HIDDEN, hW);
    aggregate_kernel<<<agg_blocks, B, 0, stream>>>(hW, rowoff, csr_src, csr_w, deg,
                                                   convB + 4 * HIDDEN, x5, 1);
    // L5: x4+x5 -> x6      (no relu)
    gemm_wmma_kernel<true><<<gemm_blocks, B, 0, stream>>>(x4, x5, convW + 5 * HIDDEN * HIDDEN, hW);
    aggregate_kernel<<<agg_blocks, B, 0, stream>>>(hW, rowoff, csr_src, csr_w, deg,
                                                   convB + 5 * HIDDEN, x6, 0);

    counts_kernel<<<(N_NODES + B - 1) / B, B, 0, stream>>>(batch, counts);
    pool_kernel<<<agg_blocks, B, 0, stream>>>(x6, batch, pooled);
    final_kernel<<<(N_GRAPHS * N_CLASSES + B - 1) / B, B, 0, stream>>>(
        pooled, counts, linW, linb, out);
}